// proj_frustrum_ego_69432441307692
// MI455X (gfx1250) — compile-verified
//
#include <hip/hip_runtime.h>
#include <hip/hip_bf16.h>

typedef __attribute__((ext_vector_type(2))) float v2f;
typedef __attribute__((ext_vector_type(8))) float v8f;

#define B_      2
#define N_      6
#define H_      64
#define W_      176
#define D_      80
#define P_      (H_ * W_ * D_)      // 901120
#define XB_     200
#define YB_     200
#define ZB_     16
#define FILL_   9
#define D_START_ 2.0f
#define D_STEP_  0.5f
// 1/0.4 and -min/0.4 biases (exactly representable)
#define INV_VOX_ 2.5f
#define X_BIAS_  100.0f   // -(-40.0)*2.5
#define Y_BIAS_  100.0f
#define Z_BIAS_  2.5f     // -(-1.0)*2.5

#define PIX_TILES_   (H_ * W_ / 16)   // 704 row-aligned 16-pixel tiles
#define D_CHUNKS_    5                // 80 depths = 5 chunks of 16
#define WAVES_PER_BLOCK_ 8

// ---------------------------------------------------------------------------
// 3x3 inverse via adjugate (per-camera, runs on 12 threads total)
// ---------------------------------------------------------------------------
__device__ inline void inv3(const float* m, float* o) {
    float a = m[0], b = m[1], c = m[2];
    float d = m[3], e = m[4], f = m[5];
    float g = m[6], h = m[7], i = m[8];
    float A  =  (e * i - f * h);
    float Bc = -(d * i - f * g);
    float C  =  (d * h - e * g);
    float det = a * A + b * Bc + c * C;
    float r = 1.0f / det;
    o[0] = A * r;   o[1] = -(b * i - c * h) * r;  o[2] =  (b * f - c * e) * r;
    o[3] = Bc * r;  o[4] =  (a * i - c * g) * r;  o[5] = -(a * f - c * d) * r;
    o[6] = C * r;   o[7] = -(a * h - b * g) * r;  o[8] =  (a * e - b * d) * r;
}

// ---------------------------------------------------------------------------
// Per-(b,n) combined transform:  ego = depth * (A @ [x,y,1] + bvec) + t
// T layout (16 floats per camera): rows 0..2 = {A[r][0..2], bvec[r]}, then t
// ---------------------------------------------------------------------------
__global__ void make_T_kernel(const float* __restrict__ post_rot,
                              const float* __restrict__ post_trans,
                              const float* __restrict__ cam2img,
                              const float* __restrict__ cam2egos,
                              float* __restrict__ T) {
    int i = blockIdx.x * blockDim.x + threadIdx.x;
    if (i >= B_ * N_) return;
    float M1[9], M2[9];
    inv3(post_rot + i * 9, M1);
    inv3(cam2img + i * 9, M2);
    const float* E = cam2egos + i * 16;

    float M21[9];
    #pragma unroll
    for (int r = 0; r < 3; ++r)
        #pragma unroll
        for (int c = 0; c < 3; ++c)
            M21[r * 3 + c] = M2[r * 3 + 0] * M1[0 * 3 + c]
                           + M2[r * 3 + 1] * M1[1 * 3 + c]
                           + M2[r * 3 + 2] * M1[2 * 3 + c];
    float A[9];
    #pragma unroll
    for (int r = 0; r < 3; ++r)
        #pragma unroll
        for (int c = 0; c < 3; ++c)
            A[r * 3 + c] = E[r * 4 + 0] * M21[0 * 3 + c]
                         + E[r * 4 + 1] * M21[1 * 3 + c]
                         + E[r * 4 + 2] * M21[2 * 3 + c];

    const float* pt = post_trans + i * 3;
    float* o = T + i * 16;
    #pragma unroll
    for (int r = 0; r < 3; ++r) {
        float bv = -(A[r * 3 + 0] * pt[0] + A[r * 3 + 1] * pt[1] + A[r * 3 + 2] * pt[2]);
        o[r * 4 + 0] = A[r * 3 + 0];
        o[r * 4 + 1] = A[r * 3 + 1];
        o[r * 4 + 2] = A[r * 3 + 2];
        o[r * 4 + 3] = bv;
    }
    o[12] = E[3];  o[13] = E[7];  o[14] = E[11];  o[15] = 0.0f;
}

__global__ void fill_occ_kernel(int* __restrict__ occ, int n) {
    int i = blockIdx.x * blockDim.x + threadIdx.x;
    if (i < n) occ[i] = FILL_;
}

// ---------------------------------------------------------------------------
// Main kernel. One V_WMMA_F32_16X16X4_F32 per wave transforms 16 PIXELS for
// two cameras at once (A rows 0-2 = cam n0's [A|b], rows 8-10 = cam n1's).
// B columns = [x, y, 1, 1]^T of 16 row-adjacent pixels. Each lane then owns
// one (pixel, camera) ray and sweeps a 16-depth chunk, reusing q = A p + b:
//     ego = depth * q + t   ->  voxelize with one FMA per axis  ->  scatter.
// ---------------------------------------------------------------------------
__global__ __launch_bounds__(256) void proj_scatter_kernel(
        const float* __restrict__ depth_map,
        const int*   __restrict__ sem_map,
        const float* __restrict__ T,
        int* __restrict__ occ) {
    const int lane = threadIdx.x & 31;
    const int wave = threadIdx.x >> 5;
    const int pt   = blockIdx.x * WAVES_PER_BLOCK_ + wave;   // 16-pixel tile id
    const int pair = blockIdx.y;                              // camera pair 0..2
    const int b    = blockIdx.z / D_CHUNKS_;                  // batch
    const int chnk = blockIdx.z % D_CHUNKS_;                  // depth chunk
    const int camBase = pair * 2;

    const int m  = lane & 15;    // A row / pixel column
    const int hi = lane >> 4;    // 0: K=0,1 & cam n0 ; 1: K=2,3 & cam n1
    const int k0 = hi * 2;

    // --- A operand ----------------------------------------------------------
    const float* T0 = T + (b * N_ + camBase) * 16;
    const float* T1 = T0 + 16;
    float a0 = 0.0f, a1 = 0.0f;
    if (m < 3)                    { a0 = T0[m * 4 + k0];       a1 = T0[m * 4 + k0 + 1]; }
    else if (m >= 8 && m < 11)    { a0 = T1[(m - 8) * 4 + k0]; a1 = T1[(m - 8) * 4 + k0 + 1]; }
    v2f Av; Av.x = a0; Av.y = a1;

    const float* Tc = hi ? T1 : T0;
    const float tx = Tc[12], ty = Tc[13], tz = Tc[14];
    const int cam = camBase + hi;

    // --- this lane's pixel (16 consecutive pixels in one image row) ---------
    const int pix0 = pt * 16;           // W = 176 = 11*16: tiles never cross rows
    const int h    = pix0 / W_;
    const int w    = (pix0 - h * W_) + m;

    // --- B operand: columns = [x, y, 1, 1]^T --------------------------------
    v2f Bv;
    if (hi == 0) { Bv.x = (float)w; Bv.y = (float)h; }   // K=0 (x), K=1 (y)
    else         { Bv.x = 1.0f;     Bv.y = 1.0f;     }   // K=2, K=3 (ones)

    v8f Cv = {};
    v8f Dv = __builtin_amdgcn_wmma_f32_16x16x4_f32(
        false, Av, false, Bv, (short)0, Cv, false, false);

    // lanes 0-15 read rows 0..2 (cam n0); lanes 16-31 read rows 8..10 (cam n1)
    const float qx = Dv[0], qy = Dv[1], qz = Dv[2];

    // --- loop-invariant per-ray data ----------------------------------------
    const int   pix  = ((b * N_ + cam) * H_ + h) * W_ + w;
    const float dobs = depth_map[pix];
    const int   sm   = sem_map[pix];

    // --- sweep 16 depths of this ray ----------------------------------------
    const int ddBase = chnk * 16;
    #pragma unroll
    for (int j = 0; j < 16; ++j) {
        const float depth = D_START_ + D_STEP_ * (float)(ddBase + j);
        const float ex = fmaf(depth, qx, tx);
        const float ey = fmaf(depth, qy, ty);
        const float ez = fmaf(depth, qz, tz);
        const int xi = (int)fmaf(ex, INV_VOX_, X_BIAS_);
        const int yi = (int)fmaf(ey, INV_VOX_, Y_BIAS_);
        const int zi = (int)fmaf(ez, INV_VOX_, Z_BIAS_);
        const int sem = (dobs <= depth) ? sm : FILL_;
        if (xi >= 0 && xi < XB_ && yi >= 0 && yi < YB_ && zi >= 0 && zi < ZB_) {
            occ[((b * XB_ + xi) * YB_ + yi) * ZB_ + zi] = sem;
        }
    }
}

// ---------------------------------------------------------------------------
extern "C" void kernel_launch(void* const* d_in, const int* in_sizes, int n_in,
                              void* d_out, int out_size, void* d_ws, size_t ws_size,
                              hipStream_t stream) {
    const float* depth_map  = (const float*)d_in[0];   // (B,N,H,W) f32
    const int*   sem_map    = (const int*)  d_in[1];   // (B,N,H,W) i32
    const float* post_rot   = (const float*)d_in[2];   // (B,N,3,3)
    const float* post_trans = (const float*)d_in[3];   // (B,N,3)
    const float* cam2img    = (const float*)d_in[4];   // (B,N,3,3)
    const float* cam2egos   = (const float*)d_in[5];   // (B,N,4,4)

    int*   occ = (int*)d_out;                          // (B,XB,YB,ZB) i32
    float* T   = (float*)d_ws;                         // 12 cameras * 16 floats

    const int nvox = B_ * XB_ * YB_ * ZB_;
    fill_occ_kernel<<<(nvox + 255) / 256, 256, 0, stream>>>(occ, nvox);
    make_T_kernel<<<1, 32, 0, stream>>>(post_rot, post_trans, cam2img, cam2egos, T);

    // grid: x = pixel tiles / waves-per-block, y = camera pairs, z = B * depth-chunks
    dim3 grid(PIX_TILES_ / WAVES_PER_BLOCK_, N_ / 2, B_ * D_CHUNKS_);  // (88, 3, 10)
    proj_scatter_kernel<<<grid, 256, 0, stream>>>(depth_map, sem_map, T, occ);
}